// MahaClassifier_23871428231951
// MI455X (gfx1250) — compile-verified
//
#include <hip/hip_runtime.h>
#include <math.h>

#define EPS 1e-12f

typedef __attribute__((ext_vector_type(2))) float v2f;
typedef __attribute__((ext_vector_type(8))) float v8f;

// ---------------------------------------------------------------------------
// Stage 1: column sum-of-squares of weight2 (axis-0 norms), atomic accumulate.
// ---------------------------------------------------------------------------
__global__ void colsq_kernel(const float* __restrict__ w2, float* __restrict__ colsq,
                             int D, int rowsPerBlk) {
    int j  = blockIdx.x * blockDim.x + threadIdx.x;   // column
    int i0 = blockIdx.y * rowsPerBlk;
    float acc = 0.0f;
    for (int i = i0; i < i0 + rowsPerBlk; ++i) {
        float v = w2[(size_t)i * D + j];
        acc += v * v;
    }
    atomicAdd(&colsq[j], acc);
}

__global__ void finalize_inv_kernel(float* __restrict__ colsq, int D) {
    int j = blockIdx.x * blockDim.x + threadIdx.x;
    if (j < D) colsq[j] = 1.0f / fmaxf(sqrtf(colsq[j]), EPS);
}

// ---------------------------------------------------------------------------
// Stage 2: s[d] = sum_j w2[d,j] * invcol[j]   (row-sum of normalized columns)
// ---------------------------------------------------------------------------
__global__ void rowsum_s_kernel(const float* __restrict__ w2, const float* __restrict__ inv,
                                float* __restrict__ s, int D) {
    __shared__ float red[256];
    int d = blockIdx.x;
    const float* row = w2 + (size_t)d * D;
    float acc = 0.0f;
    for (int j = threadIdx.x; j < D; j += blockDim.x)
        acc += row[j] * inv[j];
    red[threadIdx.x] = acc;
    __syncthreads();
    for (int off = 128; off > 0; off >>= 1) {
        if ((int)threadIdx.x < off) red[threadIdx.x] += red[threadIdx.x + off];
        __syncthreads();
    }
    if (threadIdx.x == 0) s[d] = red[0];
}

// ---------------------------------------------------------------------------
// Stage 3: r[i] = dot(row_i, s) / max(||row_i||, eps)   (used for xs and ms)
// float4 loads: one block per row, coalesced.
// ---------------------------------------------------------------------------
__global__ void dot_over_norm_kernel(const float* __restrict__ mat, const float* __restrict__ s,
                                     float* __restrict__ out, int D) {
    __shared__ float sdot[256];
    __shared__ float ssq[256];
    int row = blockIdx.x;
    const float4* r4 = (const float4*)(mat + (size_t)row * D);
    const float4* s4 = (const float4*)s;
    float dot = 0.0f, sq = 0.0f;
    int n4 = D >> 2;
    for (int j = threadIdx.x; j < n4; j += blockDim.x) {
        float4 a = r4[j];
        float4 b = s4[j];
        dot += a.x * b.x + a.y * b.y + a.z * b.z + a.w * b.w;
        sq  += a.x * a.x + a.y * a.y + a.z * a.z + a.w * a.w;
    }
    sdot[threadIdx.x] = dot;
    ssq[threadIdx.x]  = sq;
    __syncthreads();
    for (int off = 128; off > 0; off >>= 1) {
        if ((int)threadIdx.x < off) {
            sdot[threadIdx.x] += sdot[threadIdx.x + off];
            ssq[threadIdx.x]  += ssq[threadIdx.x + off];
        }
        __syncthreads();
    }
    if (threadIdx.x == 0)
        out[row] = sdot[0] / fmaxf(sqrtf(ssq[0]), EPS);
}

// ---------------------------------------------------------------------------
// Stage 4 (dominant, 512 MB of writes): out[b,c] = ms[c] - xs[b] via
// V_WMMA_F32_16X16X4_F32 rank-2 outer product:
//   A[m][0] = -xs[m], A[m][1] = 1   (K=2,3 zero)
//   B[0][n] = 1,      B[1][n] = ms[n]
//   D[m][n] = ms[n] - xs[m]
// A layout (16x4 f32): VGPR0 = K=0 (lanes 0-15) / K=2 (lanes 16-31),
//                      VGPR1 = K=1 / K=3.  B (4x16) analogous.
// C/D (16x16 f32): VGPR r -> M=r (lanes 0-15), M=r+8 (lanes 16-31), N=lane&15.
// Each wave reuses A across NT=8 column tiles (a 16x128 strip).
// ---------------------------------------------------------------------------
#define NT 8

__global__ void maha_out_wmma_kernel(const float* __restrict__ xs, const float* __restrict__ ms,
                                     float* __restrict__ out, int Brows, int Ccols) {
    const int lane = threadIdx.x & 31;
    const int wave = threadIdx.x >> 5;
    const int half = lane >> 4;       // 0: lanes 0-15, 1: lanes 16-31
    const int m    = lane & 15;

    const int rowBase  = blockIdx.y * 16;
    const int colBase0 = (blockIdx.x * 8 + wave) * (16 * NT);

    // Build A once per wave (all 32 lanes execute; zero the K=2/3 half).
    float xsv = -xs[rowBase + m];
    v2f a;
    a.x = (half == 0) ? xsv  : 0.0f;   // K=0 row / K=2 row (zero)
    a.y = (half == 0) ? 1.0f : 0.0f;   // K=1 row / K=3 row (zero)

    for (int t = 0; t < NT; ++t) {
        const int colBase = colBase0 + t * 16;
        float msv = ms[colBase + m];
        v2f b;
        b.x = (half == 0) ? 1.0f : 0.0f;   // B[0][n] = 1   / B[2][n] = 0
        b.y = (half == 0) ? msv  : 0.0f;   // B[1][n] = ms  / B[3][n] = 0
        v8f c = {};
        c = __builtin_amdgcn_wmma_f32_16x16x4_f32(
                /*neg_a=*/false, a, /*neg_b=*/false, b,
                /*c_mod=*/(short)0, c, /*reuse_a=*/false, /*reuse_b=*/false);

        const int row0 = rowBase + (half ? 8 : 0);
        size_t base = (size_t)row0 * (size_t)Ccols + (size_t)(colBase + m);
#pragma unroll
        for (int r = 0; r < 8; ++r)
            out[base + (size_t)r * (size_t)Ccols] = c[r];
    }
}

// ---------------------------------------------------------------------------
extern "C" void kernel_launch(void* const* d_in, const int* in_sizes, int n_in,
                              void* d_out, int out_size, void* d_ws, size_t ws_size,
                              hipStream_t stream) {
    const float* x  = (const float*)d_in[0];   // [B, D]
    const float* w  = (const float*)d_in[1];   // [C, D]
    const float* w2 = (const float*)d_in[2];   // [D, D]
    float* out = (float*)d_out;                // [B, C]

    // Derive shapes: in_sizes[2] = D*D
    int D = (int)(sqrt((double)in_sizes[2]) + 0.5);
    int B = in_sizes[0] / D;
    int C = in_sizes[1] / D;

    // Workspace layout (floats): inv[D] | s[D] | xs[B] | ms[C]
    float* wsf = (float*)d_ws;
    float* inv = wsf;
    float* s   = wsf + D;
    float* xs  = wsf + 2 * D;
    float* ms  = wsf + 2 * D + B;

    dim3 blk(256);

    // Stage 1: column norms of weight2 -> 1/max(norm, eps)
    hipMemsetAsync(inv, 0, (size_t)D * sizeof(float), stream);
    colsq_kernel<<<dim3(D / 256, 16), blk, 0, stream>>>(w2, inv, D, D / 16);
    finalize_inv_kernel<<<(D + 255) / 256, blk, 0, stream>>>(inv, D);

    // Stage 2: s = row-sums of column-normalized weight2
    rowsum_s_kernel<<<D, blk, 0, stream>>>(w2, inv, s, D);

    // Stage 3: xs = (x @ s) / row-norms(x);  ms = (weight @ s) / row-norms(weight)
    dot_over_norm_kernel<<<B, blk, 0, stream>>>(x, s, xs, D);
    dot_over_norm_kernel<<<C, blk, 0, stream>>>(w, s, ms, D);

    // Stage 4: out[b,c] = ms[c] - xs[b] via WMMA outer product (write-bound)
    maha_out_wmma_kernel<<<dim3(C / (8 * 16 * NT), B / 16), blk, 0, stream>>>(xs, ms, out, B, C);
}